// SwitchFeedForward_3186865734109
// MI455X (gfx1250) — compile-verified
//
#include <hip/hip_runtime.h>
#include <math.h>

#define NE 8
#define DM 1024
#define DF 4096
#define NT 8192          // 4 * 2048 tokens
#define BM 64            // tokens per FFN block (4 M-tiles)
#define MT 4             // M-tiles per block
#define BF 128           // FF chunk per iteration (8 waves x 16 cols)
#define XROW (DM + 4)    // padded LDS row stride (floats): bank-conflict free
#define HROW (BF + 4)

typedef __attribute__((ext_vector_type(2))) float v2f;
typedef __attribute__((ext_vector_type(8))) float v8f;
typedef __attribute__((__vector_size__(4 * sizeof(int)))) int vsi4;
typedef __attribute__((address_space(1))) vsi4 g_vsi4;  // global (device) AS
typedef __attribute__((address_space(3))) vsi4 l_vsi4;  // LDS AS

#ifndef __has_builtin
#define __has_builtin(x) 0
#endif

static __device__ __forceinline__ v8f wmma_f32(v2f a, v2f b, v8f c) {
  // D = A(16x4 f32) * B(4x16 f32) + C(16x16 f32)
  return __builtin_amdgcn_wmma_f32_16x16x4_f32(false, a, false, b, (short)0, c,
                                               false, false);
}

// ---------------- Router: softmax + fixed-expert-0 + top1-of-rest, compaction
__global__ __launch_bounds__(256) void moe_router(
    const float* __restrict__ x, const float* __restrict__ Wr,
    int* __restrict__ counts, int* __restrict__ etok, float* __restrict__ egate) {
  __shared__ float wr[NE * DM];  // 32 KB
  for (int i = threadIdx.x; i < NE * DM; i += 256) wr[i] = Wr[i];
  __syncthreads();

  const int lane = threadIdx.x & 31;
  const int wave = threadIdx.x >> 5;
  const int t = blockIdx.x * 8 + wave;  // one token per wave
  if (t >= NT) return;
  const float* xr = x + (size_t)t * DM;

  float acc[NE];
#pragma unroll
  for (int e = 0; e < NE; ++e) acc[e] = 0.f;
  for (int d = lane; d < DM; d += 32) {
    float xv = xr[d];
#pragma unroll
    for (int e = 0; e < NE; ++e) acc[e] += xv * wr[e * DM + d];
  }
#pragma unroll
  for (int e = 0; e < NE; ++e) {
    float v = acc[e];
#pragma unroll
    for (int off = 16; off >= 1; off >>= 1) v += __shfl_xor(v, off, 32);
    acc[e] = v;
  }
  if (lane == 0) {
    float mx = acc[0];
#pragma unroll
    for (int e = 1; e < NE; ++e) mx = fmaxf(mx, acc[e]);
    float ex[NE], s = 0.f;
#pragma unroll
    for (int e = 0; e < NE; ++e) { ex[e] = __expf(acc[e] - mx); s += ex[e]; }
    float p0 = ex[0] / s;
    int best = 1;
    float bp = ex[1];
#pragma unroll
    for (int e = 2; e < NE; ++e)
      if (ex[e] > bp) { bp = ex[e]; best = e; }
    bp /= s;
    float denom = fmaxf(p0 + bp, 1e-9f);
    etok[t] = t;                 // expert 0: identity slot (always selected)
    egate[t] = p0 / denom;
    int slot = atomicAdd(&counts[best], 1);
    etok[best * NT + slot] = t;
    egate[best * NT + slot] = bp / denom;
  }
}

// ---------------- Fused expert FFN over compacted token lists
__global__ __launch_bounds__(256) void moe_ffn(
    const float* __restrict__ x, const float* __restrict__ W1,
    const float* __restrict__ W2, const float* __restrict__ W3,
    const int* __restrict__ counts, const int* __restrict__ etok,
    const float* __restrict__ egate, float* __restrict__ out) {
  const int e = blockIdx.y;
  const int count = (e == 0) ? NT : counts[e];
  const int tile0 = blockIdx.x * BM;
  if (tile0 >= count) return;

  extern __shared__ char smem[];
  float* xs = (float*)smem;                // BM x XROW
  float* hs = xs + BM * XROW;              // BM x HROW
  int* tok = (int*)(hs + BM * HROW);       // BM
  float* gt = (float*)(tok + BM);          // BM

  const int tid = threadIdx.x;
  const int lane = tid & 31;
  const int wave = tid >> 5;
  const int nlo = lane & 15;    // M (for A) / N (for B,C) within a 16x16 tile
  const int khalf = lane >> 4;  // selects K {0,1} vs {2,3} halves

  if (tid < BM) {
    int r = tile0 + tid;
    bool ok = (r < count);
    tok[tid] = ok ? etok[e * NT + r] : 0;
    gt[tid] = ok ? egate[e * NT + r] : 0.f;
  }
  __syncthreads();

  // Stage gathered x rows into LDS via gfx1250 async global->LDS DMA
  // (ASYNCcnt-tracked, no VGPR round trip); manual staging as fallback.
  // AS1/AS3 pointers formed by integer round-trip: generic LDS addresses are
  // {SHARED_BASE, offset[31:0]} and hardware truncates to addr[31:0].
#if __has_builtin(__builtin_amdgcn_global_load_async_to_lds_b128)
  {
    for (int r = wave; r < BM; r += 8) {
      g_vsi4* src = (g_vsi4*)(uintptr_t)(x + (size_t)tok[r] * DM);
      l_vsi4* dst = (l_vsi4*)(unsigned)(uintptr_t)(xs + (size_t)r * XROW);
      // 1024 floats = 256 x 16B chunks, 32 lanes per wave
      for (int c = lane; c < 256; c += 32)
        __builtin_amdgcn_global_load_async_to_lds_b128(src + c, dst + c, 0, 0);
    }
    asm volatile("s_wait_asynccnt 0" ::: "memory");
  }
#else
  for (int r = wave; r < BM; r += 8) {
    const float* src = x + (size_t)tok[r] * DM;
    for (int d = lane; d < DM; d += 32) xs[r * XROW + d] = src[d];
  }
#endif
  __syncthreads();

  const float* W1e = W1 + (size_t)e * DF * DM;
  const float* W3e = W3 + (size_t)e * DF * DM;
  const float* W2e = W2 + (size_t)e * DM * DF;

  v8f oacc[MT][8];  // persistent accumulators: 64 tokens x 128 cols/wave
#pragma unroll
  for (int i = 0; i < MT; ++i)
#pragma unroll
    for (int j = 0; j < 8; ++j) oacc[i][j] = {};

  for (int f0 = 0; f0 < DF; f0 += BF) {
    // ---- phase 1: hs = silu(x*W1^T) * (x*W3^T) for this 128-wide FF chunk.
    // Each wave owns 16 FF columns and computes all 4 token M-tiles from one
    // B-fragment load: 8 WMMAs per W1/W3 load pair.
    {
      const int frow = f0 + wave * 16 + nlo;  // B column n -> weight row
      const float* w1r = W1e + (size_t)frow * DM;
      const float* w3r = W3e + (size_t)frow * DM;
      const float* ar[MT];
#pragma unroll
      for (int mt = 0; mt < MT; ++mt)
        ar[mt] = xs + (size_t)(mt * 16 + nlo) * XROW;
      if (f0 + BF < DF) {  // prefetch next chunk's weight rows into L2
        __builtin_prefetch(W1e + (size_t)(frow + BF) * DM, 0, 1);
        __builtin_prefetch(W3e + (size_t)(frow + BF) * DM, 0, 1);
      }
      v8f h1[MT] = {{}, {}, {}, {}}, h3[MT] = {{}, {}, {}, {}};
#pragma unroll 2
      for (int k = 0; k < DM; k += 4) {
        int kk = k + 2 * khalf;
        v2f b1 = *(const v2f*)(w1r + kk);
        v2f b3 = *(const v2f*)(w3r + kk);
#pragma unroll
        for (int mt = 0; mt < MT; ++mt) {
          v2f a = *(const v2f*)(ar[mt] + kk);
          h1[mt] = wmma_f32(a, b1, h1[mt]);
          h3[mt] = wmma_f32(a, b3, h3[mt]);
        }
      }
#pragma unroll
      for (int mt = 0; mt < MT; ++mt) {
#pragma unroll
        for (int i = 0; i < 8; ++i) {  // C/D layout: VGPR i -> row i + 8*khalf
          int row = mt * 16 + i + 8 * khalf;
          float s = h1[mt][i];
          // silu(s) * h3 with fast v_rcp_f32 (avoids IEEE div fixup chain)
          float h = s * __builtin_amdgcn_rcpf(1.f + __expf(-s)) * h3[mt][i];
          hs[row * HROW + wave * 16 + nlo] = h;
        }
      }
    }
    __syncthreads();
    // ---- phase 2: oacc += h_chunk @ W2_chunk^T (each wave: 128 out columns).
    // One W2 fragment feeds all 4 M-tiles: 32 WMMAs per 8 B loads per k-step.
    {
      const int cb = wave * 128;
      const float* ar[MT];
#pragma unroll
      for (int mt = 0; mt < MT; ++mt)
        ar[mt] = hs + (size_t)(mt * 16 + nlo) * HROW;
      for (int k = 0; k < BF; k += 4) {
        int kk = k + 2 * khalf;
        v2f a[MT];
#pragma unroll
        for (int mt = 0; mt < MT; ++mt) a[mt] = *(const v2f*)(ar[mt] + kk);
#pragma unroll
        for (int nt2 = 0; nt2 < 8; ++nt2) {
          const float* w2r = W2e + (size_t)(cb + nt2 * 16 + nlo) * DF + f0;
          v2f bv = *(const v2f*)(w2r + kk);
#pragma unroll
          for (int mt = 0; mt < MT; ++mt)
            oacc[mt][nt2] = wmma_f32(a[mt], bv, oacc[mt][nt2]);
        }
      }
    }
    __syncthreads();
  }

  // ---- gated writeback: exactly 2 expert contributions per token -> atomicAdd
  const int cb = wave * 128;
#pragma unroll
  for (int mt2 = 0; mt2 < MT; ++mt2) {
#pragma unroll
    for (int i = 0; i < 8; ++i) {
      int row = mt2 * 16 + i + 8 * khalf;
      int t = tok[row];
      float g = gt[row];
      float* orow = out + (size_t)t * DM + cb;
#pragma unroll
      for (int nt2 = 0; nt2 < 8; ++nt2)
        atomicAdd(orow + nt2 * 16 + nlo, g * oacc[mt2][nt2][i]);
    }
  }
}

extern "C" void kernel_launch(void* const* d_in, const int* in_sizes, int n_in,
                              void* d_out, int out_size, void* d_ws,
                              size_t ws_size, hipStream_t stream) {
  const float* x = (const float*)d_in[0];
  const float* Wr = (const float*)d_in[1];
  const float* W1 = (const float*)d_in[2];
  const float* W2 = (const float*)d_in[3];
  const float* W3 = (const float*)d_in[4];
  float* out = (float*)d_out;

  char* ws = (char*)d_ws;
  int* counts = (int*)ws;                                   // 8 ints (256B)
  int* etok = (int*)(ws + 256);                             // NE*NT ints
  float* egate = (float*)(ws + 256 + (size_t)NE * NT * 4);  // NE*NT floats

  (void)hipMemsetAsync(counts, 0, 256, stream);
  (void)hipMemsetAsync(out, 0, (size_t)NT * DM * sizeof(float), stream);

  moe_router<<<NT / 8, 256, 0, stream>>>(x, Wr, counts, etok, egate);

  dim3 grid(NT / BM, NE);
  size_t smem = (size_t)(BM * XROW + BM * HROW) * sizeof(float) + BM * 8;
  moe_ffn<<<grid, 256, smem, stream>>>(x, W1, W2, W3, counts, etok, egate, out);
}